// HypergraphOutputNormalByRow_28286654612010
// MI455X (gfx1250) — compile-verified
//
#include <hip/hip_runtime.h>

// segment_max (scatter-max) + gather + residual + [N,2,D] pack for
// N=1e6, D=128, R=5e4 on MI455X (gfx1250, wave32).
//
// Memory-bound op: ~2.05 GB mandatory HBM traffic -> ~88us floor @ 23.3 TB/s.
// No contraction -> WMMA is not applicable; the CDNA5 features that matter:
//  - async global->LDS b128 staging (ASYNCcnt / s_wait_asynccnt) for the
//    streaming feat re-read (double-buffered, self-consumed, barrier-free)
//  - native global_atomic_max_u32 into the L2-resident 25.6MB segment table
//    (order-preserving u32 key transform makes float max exact + deterministic)
//  - NT temporal hints on the write-once 1GB output / stream-once feat read
//    so they don't evict the hot segment table from the 192MB L2.
//
// k_emit is defined FIRST so the disasm snippet shows the async-tensor path.

typedef __attribute__((ext_vector_type(4))) float    v4f;
typedef __attribute__((ext_vector_type(4))) unsigned v4u;
typedef __attribute__((ext_vector_type(4))) int      v4i;

#define THREADS 256

#if defined(__has_builtin)
#  if __has_builtin(__builtin_amdgcn_global_load_async_to_lds_b128) && \
      __has_builtin(__builtin_amdgcn_s_wait_asynccnt)
#    define USE_ASYNC_LDS 1
#  endif
#endif

#if defined(USE_ASYNC_LDS)
typedef __attribute__((address_space(1))) v4i* gp128_t;   // global int4*
typedef __attribute__((address_space(3))) v4i* lp128_t;   // LDS int4*
#endif

// Order-preserving monotone map float -> u32 so native atomicMax(u32) computes
// float max exactly (inputs are normal-distributed, no NaNs).
__device__ __forceinline__ unsigned enc_f32(float f) {
  unsigned b = __float_as_uint(f);
  return (b & 0x80000000u) ? ~b : (b | 0x80000000u);
}
__device__ __forceinline__ float dec_f32(unsigned k) {
  unsigned b = (k & 0x80000000u) ? (k & 0x7fffffffu) : ~k;
  return __uint_as_float(b);
}

// Pass 2: re-stream feat through LDS with double-buffered async global->LDS
// copies; decode the gathered segment max, emit residual + copy as two NT
// b128 stores per thread. Each thread consumes only its own LDS slot, so the
// only synchronization needed is s_wait_asynccnt (no workgroup barriers).
template <int TPB_TILES>
__global__ void k_emit(const v4f* __restrict__ feat4,
                       const int* __restrict__ row_ids,
                       const v4u* __restrict__ seg4,
                       v4f* __restrict__ out4,
                       int total4) {
  __shared__ v4f buf[2][THREADS];      // 8 KB double buffer
  const int tid  = threadIdx.x;
  const int base = blockIdx.x * (TPB_TILES * THREADS);

  auto stage = [&](int t) {
    int idx = base + t * THREADS + tid;
    if (idx >= total4) idx = total4 - 1;     // clamp: address always legal
#if defined(USE_ASYNC_LDS)
    __builtin_amdgcn_global_load_async_to_lds_b128(
        (gp128_t)(feat4 + idx),
        (lp128_t)&buf[t & 1][tid],
        /*offset=*/0, /*cpol=*/0);
#else
    buf[t & 1][tid] = feat4[idx];
#endif
  };

  stage(0);
  for (int t = 0; t < TPB_TILES; ++t) {
    if (t + 1 < TPB_TILES) {
      stage(t + 1);                          // keep next tile in flight
#if defined(USE_ASYNC_LDS)
      __builtin_amdgcn_s_wait_asynccnt(1);   // tile t has landed in LDS
#endif
    } else {
#if defined(USE_ASYNC_LDS)
      __builtin_amdgcn_s_wait_asynccnt(0);
#endif
    }

    int idx = base + t * THREADS + tid;
    if (idx < total4) {
      v4f f = buf[t & 1][tid];               // ds_load_b128 (own slot)
      int n = idx >> 5;                      // D == 128 -> D4 == 32
      int j = idx & 31;
      int r = row_ids[n];                    // broadcast within wave32
      v4u k = seg4[(r << 5) + j];            // gather, L2-resident (RT hint)
      v4f m;
      m.x = dec_f32(k.x); m.y = dec_f32(k.y);
      m.z = dec_f32(k.z); m.w = dec_f32(k.w);
      int ob = (n << 6) + j;                 // out row = 64 float4: [res|feat]
      __builtin_nontemporal_store(m - f, out4 + ob);        // NT b128 store
      __builtin_nontemporal_store(f,     out4 + ob + 32);   // NT b128 store
    }
  }
}

// Pass 1: scatter-max. One thread = one float4 of one row; 32 consecutive
// lanes share n -> row_ids[n] load broadcasts within the wave32.
// 4x global_atomic_max_u32 (no-return) per thread into the L2-resident table.
__global__ void k_segmax(const v4f* __restrict__ feat4,
                         const int* __restrict__ row_ids,
                         unsigned* __restrict__ seg,
                         int total4) {
  int idx = blockIdx.x * THREADS + threadIdx.x;
  if (idx >= total4) return;
  int n = idx >> 5;
  int j = idx & 31;
  int r = row_ids[n];
  v4f f = __builtin_nontemporal_load(feat4 + idx);  // NT b128 stream
  unsigned* p = seg + (((r << 5) + j) << 2);
  atomicMax(p + 0, enc_f32(f.x));
  atomicMax(p + 1, enc_f32(f.y));
  atomicMax(p + 2, enc_f32(f.z));
  atomicMax(p + 3, enc_f32(f.w));
}

// Pass 0: zero-init key table seg[R*D]. R lives only on the device (graph
// capture forbids a sync readback), so read it here and grid-stride.
// Key 0 sorts below every encoded real value (= -inf identity); empty
// segments are never gathered, matching jax.ops.segment_max semantics.
__global__ void k_init(unsigned* __restrict__ seg,
                       const int* __restrict__ num_rows, int D) {
  const int total  = (*num_rows) * D;                  // 6.4M, fits in int
  const int stride = gridDim.x * blockDim.x;
  for (int i = blockIdx.x * blockDim.x + threadIdx.x; i < total; i += stride)
    seg[i] = 0u;
}

extern "C" void kernel_launch(void* const* d_in, const int* in_sizes, int n_in,
                              void* d_out, int out_size, void* d_ws, size_t ws_size,
                              hipStream_t stream) {
  const float* feat     = (const float*)d_in[0];
  const int*   row_ids  = (const int*)d_in[1];
  const int*   num_rows = (const int*)d_in[2];   // device scalar
  const int N      = in_sizes[1];
  const int D      = in_sizes[0] / N;            // 128 per reference
  const int total4 = N * (D >> 2);               // 32M float4s
  unsigned* seg = (unsigned*)d_ws;               // R*D u32 keys (25.6 MB)

  k_init<<<2048, THREADS, 0, stream>>>(seg, num_rows, D);

  int b1 = (total4 + THREADS - 1) / THREADS;
  k_segmax<<<b1, THREADS, 0, stream>>>((const v4f*)feat, row_ids, seg, total4);

  constexpr int TPB = 16;                        // 16 tiles x 4KB per block
  int tiles = (total4 + THREADS - 1) / THREADS;
  int b2    = (tiles + TPB - 1) / TPB;
  k_emit<TPB><<<b2, THREADS, 0, stream>>>((const v4f*)feat, row_ids,
                                          (const v4u*)seg, (v4f*)d_out, total4);

  (void)n_in; (void)out_size; (void)ws_size;
}